// AxialSoftAttention_33878702030934
// MI455X (gfx1250) — compile-verified
//
#include <hip/hip_runtime.h>
#include <hip/hip_bf16.h>

// ---------- CDNA5 WMMA types / wrapper ----------
typedef float v2f __attribute__((ext_vector_type(2)));
typedef float v8f __attribute__((ext_vector_type(8)));
typedef unsigned int u32x4 __attribute__((ext_vector_type(4)));
typedef int i32x4 __attribute__((ext_vector_type(4)));
typedef int i32x8 __attribute__((ext_vector_type(8)));

__device__ __forceinline__ v8f zero8() {
  v8f z = {0.f, 0.f, 0.f, 0.f, 0.f, 0.f, 0.f, 0.f};
  return z;
}

// D = A(16x4 f32) * B(4x16 f32) + C   (V_WMMA_F32_16X16X4_F32)
__device__ __forceinline__ v8f wmma4(v2f a, v2f b, v8f c) {
  return __builtin_amdgcn_wmma_f32_16x16x4_f32(
      /*neg_a=*/false, a, /*neg_b=*/false, b,
      /*c_mod=*/(short)0, c, /*reuse_a=*/false, /*reuse_b=*/false);
}

// ---------- Tensor Data Mover: contiguous 1D global->LDS copy ----------
#if defined(__gfx1250__)
#if defined(__has_builtin)
#if __has_builtin(__builtin_amdgcn_tensor_load_to_lds)
#define USE_TDM 1
#endif
#endif
#endif

__device__ __forceinline__ void tdm_load_1d(const void* gsrc, void* ldst,
                                            unsigned nelem /* 4-byte elems, <65536 */) {
#ifdef USE_TDM
  unsigned long long ga = (unsigned long long)gsrc;
  unsigned la = (unsigned)(unsigned long long)ldst;  // LDS offset = addr[31:0]
  // D# group 0: count=1 | lds_addr | global_addr[56:0] | type=2
  u32x4 g0;
  g0[0] = 1u;
  g0[1] = la;
  g0[2] = (unsigned)ga;
  g0[3] = ((unsigned)(ga >> 32) & 0x01FFFFFFu) | 0x80000000u;
  // D# group 1: data_size=4B; tensor_dim0=tile_dim0=nelem; dim1=1; stride0=nelem
  i32x8 g1;
  g1[0] = 0x00020000;                                  // data_size=2 (4 bytes)
  g1[1] = (int)((nelem & 0xFFFFu) << 16);              // tensor_dim0[15:0]
  g1[2] = (int)(((nelem >> 16) & 0xFFFFu) | (1u << 16)); // dim0 hi | tensor_dim1=1
  g1[3] = (int)((nelem & 0xFFFFu) << 16);              // tile_dim0
  g1[4] = 1;                                           // tile_dim1=1, tile_dim2=0
  g1[5] = (int)nelem;                                  // tensor_dim0_stride
  g1[6] = 0;
  g1[7] = 0;
  i32x4 z4 = {0, 0, 0, 0};
#if __clang_major__ >= 23
  i32x8 z8 = {0, 0, 0, 0, 0, 0, 0, 0};
  __builtin_amdgcn_tensor_load_to_lds(g0, g1, z4, z4, z8, 0);
#else
  __builtin_amdgcn_tensor_load_to_lds(g0, g1, z4, z4, 0);
#endif
#endif
}

__device__ __forceinline__ void tdm_wait() {
#ifdef USE_TDM
  __builtin_amdgcn_s_wait_tensorcnt(0);
#endif
}

// ---------- problem constants ----------
#define Cc   64
#define DCc  16
#define Ff   256
#define Tt   384
#define Bb   4
#define Npix (Ff * Tt)          // 98304 pixels per batch
#define EPSc 1e-5f

// workspace layout (in floats)
#define OFF_WQ 0                 // 80*64 folded qkv+tqk weights
#define OFF_BQ 5120              // 80 biases
#define OFF_AQ 5200              // 80 alphas
#define OFF_WP 5280              // 64*16 folded proj weights
#define OFF_BP 6304              // 64
#define OFF_AP 6368              // 64
#define OFF_QF 6400
#define SZTEN  (Bb * Tt * Ff * DCc)   // 6291456 floats per tensor
#define OFF_KF (OFF_QF + SZTEN)
#define OFF_V  (OFF_KF + SZTEN)
#define OFF_QT (OFF_V + SZTEN)
#define OFF_KT (OFF_QT + SZTEN)
#define OFF_FO (OFF_KT + SZTEN)
#define OFF_TO (OFF_FO + SZTEN)

// ---------------------------------------------------------------
// Kernel 0: fold BN into weights: y = s*conv + b ; s=gamma*rsqrt(var+eps).
// The attention scale 1/sqrt(DC)=0.25 is folded into the q-row weights/biases
// (exact: PReLU is positively homogeneous), so later staging is a pure copy.
// ---------------------------------------------------------------
__global__ __launch_bounds__(128) void fold_params(
    const float* __restrict__ fqkv_w, const float* __restrict__ fg,
    const float* __restrict__ fb, const float* __restrict__ fm,
    const float* __restrict__ fv, const float* __restrict__ fa,
    const float* __restrict__ tqk_w, const float* __restrict__ tg,
    const float* __restrict__ tb, const float* __restrict__ tm,
    const float* __restrict__ tv, const float* __restrict__ ta,
    const float* __restrict__ proj_w, const float* __restrict__ pg,
    const float* __restrict__ pb, const float* __restrict__ pm,
    const float* __restrict__ pv, const float* __restrict__ pa,
    float* __restrict__ ws) {
  int t = threadIdx.x;
  float* Wq = ws + OFF_WQ;
  float* bq = ws + OFF_BQ;
  float* aq = ws + OFF_AQ;
  float* Wp = ws + OFF_WP;
  float* bp = ws + OFF_BP;
  float* ap = ws + OFF_AP;
  if (t < 48) {
    float sc = (t < 16) ? 0.25f : 1.0f;   // qf rows carry the score scale
    float s = fg[t] * rsqrtf(fv[t] + EPSc) * sc;
    for (int c = 0; c < 64; ++c) Wq[t * 64 + c] = fqkv_w[t * 64 + c] * s;
    bq[t] = sc * fb[t] - fm[t] * s;
    aq[t] = fa[t];
  } else if (t < 80) {
    int o = t - 48;
    float sc = (o < 16) ? 0.25f : 1.0f;   // qt rows carry the score scale
    float s = tg[o] * rsqrtf(tv[o] + EPSc) * sc;
    for (int c = 0; c < 64; ++c) Wq[t * 64 + c] = tqk_w[o * 64 + c] * s;
    bq[t] = sc * tb[o] - tm[o] * s;
    aq[t] = ta[o];
  }
  if (t < 64) {
    float s = pg[t] * rsqrtf(pv[t] + EPSc);
    for (int c = 0; c < 16; ++c) Wp[t * 16 + c] = proj_w[t * 16 + c] * s;
    bp[t] = pb[t] - pm[t] * s;
    ap[t] = pa[t];
  }
}

// ---------------------------------------------------------------
// Kernel A: fused QKV+TQK projection (GEMM 80x64 x pixels) + bias + PReLU.
// Writes qf/kf/v as [b][t][f][16], qt/kt as [b][f][t][16].
// grid (768, B), block 256 (8 waves), one wave = 16-pixel tile.
// ---------------------------------------------------------------
__global__ __launch_bounds__(256) void qkv_kernel(
    const float* __restrict__ x, const float* __restrict__ ws) {
  __shared__ __align__(16) float Wl[80 * 64];
  __shared__ __align__(16) float bl[80];
  __shared__ __align__(16) float al[80];
  const float* Wq = ws + OFF_WQ;
  for (int i = threadIdx.x; i < 80 * 64; i += 256) Wl[i] = Wq[i];
  if (threadIdx.x < 80) {
    bl[threadIdx.x] = ws[OFF_BQ + threadIdx.x];
    al[threadIdx.x] = ws[OFF_AQ + threadIdx.x];
  }
  __syncthreads();

  float* qf = const_cast<float*>(ws) + OFF_QF;
  float* kf = const_cast<float*>(ws) + OFF_KF;
  float* vv = const_cast<float*>(ws) + OFF_V;
  float* qt = const_cast<float*>(ws) + OFF_QT;
  float* kt = const_cast<float*>(ws) + OFF_KT;

  int wid = threadIdx.x >> 5;
  int lane = threadIdx.x & 31;
  int li = lane & 15, hi = lane >> 4, ko = hi * 2;
  int b = blockIdx.y;
  int n0 = (blockIdx.x * 8 + wid) * 16;
  const float* xb = x + (size_t)b * Cc * Npix;

  // B fragments: B[k=c][n=pixel] = x[b][c][pixel]
  v2f bf[16];
#pragma unroll
  for (int k = 0; k < 16; ++k) {
    int row = 4 * k + ko;
    bf[k].x = xb[(size_t)row * Npix + n0 + li];
    bf[k].y = xb[(size_t)(row + 1) * Npix + n0 + li];
  }

  int n = n0 + li;
  int f = n / Tt, t = n % Tt;

#pragma unroll
  for (int m = 0; m < 5; ++m) {
    v8f acc = zero8();
#pragma unroll
    for (int k = 0; k < 16; ++k) {
      v2f a;
      a.x = Wl[(16 * m + li) * 64 + 4 * k + ko];
      a.y = Wl[(16 * m + li) * 64 + 4 * k + ko + 1];
      acc = wmma4(a, bf[k], acc);
    }
    // branchless bias + PReLU epilogue
    float bi[8], aa[8];
    *(float4*)&bi[0] = *(const float4*)(bl + 16 * m + hi * 8);
    *(float4*)&bi[4] = *(const float4*)(bl + 16 * m + hi * 8 + 4);
    *(float4*)&aa[0] = *(const float4*)(al + 16 * m + hi * 8);
    *(float4*)&aa[4] = *(const float4*)(al + 16 * m + hi * 8 + 4);
    float res[8];
#pragma unroll
    for (int r = 0; r < 8; ++r) {
      float y = acc[r] + bi[r];
      res[r] = fmaxf(y, 0.f) + aa[r] * fminf(y, 0.f);
    }
    float* dst;
    size_t base;
    if (m < 3) {  // qf/kf/v : [b][t][f][c]
      dst = (m == 0) ? qf : (m == 1) ? kf : vv;
      base = (((size_t)b * Tt + t) * Ff + f) * DCc + hi * 8;
    } else {      // qt/kt : [b][f][t][c]
      dst = (m == 3) ? qt : kt;
      base = (((size_t)b * Ff + f) * Tt + t) * DCc + hi * 8;
    }
    *(float4*)(dst + base) = make_float4(res[0], res[1], res[2], res[3]);
    *(float4*)(dst + base + 4) = make_float4(res[4], res[5], res[6], res[7]);
  }
}

// ---------------------------------------------------------------
// Kernel B: F-axis attention, one workgroup per (b,t).
// S (256x256) lives entirely in LDS; Q/K/V staged by the Tensor Data Mover.
// dyn LDS: Q 4096 + K 4096 + V 4096 + S 65536 + rowsum 256 floats (305 KB).
// ---------------------------------------------------------------
__global__ __launch_bounds__(512) void fattn_kernel(const float* __restrict__ ws_in,
                                                    float* __restrict__ ws_out) {
  extern __shared__ float sm[];
  float* Qs = sm;
  float* Ks = sm + 4096;
  float* Vs = sm + 8192;
  float* Ss = sm + 12288;
  float* rs = sm + 12288 + 65536;

  const float* qf = ws_in + OFF_QF;
  const float* kf = ws_in + OFF_KF;
  const float* vv = ws_in + OFF_V;
  float* fo = ws_out + OFF_FO;

  int b = blockIdx.y, t = blockIdx.x;
  size_t base = ((size_t)b * Tt + t) * (Ff * DCc);
#ifdef USE_TDM
  if (threadIdx.x == 0) {             // wave 0 issues three async tensor DMAs
    tdm_load_1d(qf + base, Qs, Ff * DCc);
    tdm_load_1d(kf + base, Ks, Ff * DCc);
    tdm_load_1d(vv + base, Vs, Ff * DCc);
    tdm_wait();
  }
#else
  for (int i = threadIdx.x; i < 1024; i += 512) {
    ((float4*)Qs)[i] = ((const float4*)(qf + base))[i];
    ((float4*)Ks)[i] = ((const float4*)(kf + base))[i];
    ((float4*)Vs)[i] = ((const float4*)(vv + base))[i];
  }
#endif
  __syncthreads();

  int wid = threadIdx.x >> 5;
  int lane = threadIdx.x & 31;
  int li = lane & 15, hi = lane >> 4, ko = hi * 2;
  int f0 = wid * 16;

  // S = Q * K^T : per wave, one 16-row stripe, 16 column tiles
  for (int j = 0; j < 16; ++j) {
    v8f acc = zero8();
#pragma unroll
    for (int k = 0; k < 4; ++k) {
      v2f a, bb;
      a.x = Qs[(f0 + li) * 16 + 4 * k + ko];
      a.y = Qs[(f0 + li) * 16 + 4 * k + ko + 1];
      bb.x = Ks[(16 * j + li) * 16 + 4 * k + ko];
      bb.y = Ks[(16 * j + li) * 16 + 4 * k + ko + 1];
      acc = wmma4(a, bb, acc);
    }
#pragma unroll
    for (int r = 0; r < 8; ++r)
      Ss[(f0 + r + hi * 8) * 256 + 16 * j + li] = acc[r];
  }
  __syncthreads();

  // softmax rows: store un-normalized exp, keep 1/rowsum
  if (threadIdx.x < 256) {
    float4* Sr = (float4*)(Ss + threadIdx.x * 256);
    float mx = -3.4e38f;
    for (int i = 0; i < 64; ++i) {
      float4 s4 = Sr[i];
      mx = fmaxf(mx, fmaxf(fmaxf(s4.x, s4.y), fmaxf(s4.z, s4.w)));
    }
    float sum = 0.f;
    for (int i = 0; i < 64; ++i) {
      float4 s4 = Sr[i];
      s4.x = __expf(s4.x - mx); s4.y = __expf(s4.y - mx);
      s4.z = __expf(s4.z - mx); s4.w = __expf(s4.w - mx);
      sum += s4.x + s4.y + s4.z + s4.w;
      Sr[i] = s4;
    }
    rs[threadIdx.x] = 1.0f / sum;
  }
  __syncthreads();

  // O = P * V (256x256 @ 256x16); normalization folded into epilogue
  v8f acc = zero8();
  for (int k = 0; k < 64; ++k) {
    v2f a, bb;
    a.x = Ss[(f0 + li) * 256 + 4 * k + ko];
    a.y = Ss[(f0 + li) * 256 + 4 * k + ko + 1];
    bb.x = Vs[(4 * k + ko) * 16 + li];
    bb.y = Vs[(4 * k + ko + 1) * 16 + li];
    acc = wmma4(a, bb, acc);
  }
#pragma unroll
  for (int r = 0; r < 8; ++r) {
    int f = f0 + r + hi * 8;
    fo[(((size_t)b * Ff + f) * Tt + t) * DCc + li] = acc[r] * rs[f];
  }
}

// ---------------------------------------------------------------
// Kernel C: causal T-axis attention, one workgroup per (b,f,128-row stripe).
// dyn LDS: Qstripe 2048 + Kt 6144 + Fout 6144 + S 49152 + rowsum 128 (249 KB).
// ---------------------------------------------------------------
__global__ __launch_bounds__(256) void tattn_kernel(const float* __restrict__ ws_in,
                                                    float* __restrict__ ws_out) {
  extern __shared__ float sm[];
  float* Qs = sm;            // 128x16
  float* Ks = sm + 2048;     // 384x16
  float* Fs = sm + 8192;     // 384x16
  float* Ss = sm + 14336;    // 128x384
  float* rs = sm + 63488;    // 128

  const float* qt = ws_in + OFF_QT;
  const float* kt = ws_in + OFF_KT;
  const float* fo = ws_in + OFF_FO;
  float* to = ws_out + OFF_TO;

  int b = blockIdx.z, f = blockIdx.y, t0 = blockIdx.x * 128;
  size_t base = ((size_t)b * Ff + f) * (Tt * DCc);
#ifdef USE_TDM
  if (threadIdx.x == 0) {
    tdm_load_1d(qt + base + (size_t)t0 * DCc, Qs, 128 * DCc);
    tdm_load_1d(kt + base, Ks, Tt * DCc);
    tdm_load_1d(fo + base, Fs, Tt * DCc);
    tdm_wait();
  }
#else
  for (int i = threadIdx.x; i < 512; i += 256)
    ((float4*)Qs)[i] = ((const float4*)(qt + base + (size_t)t0 * DCc))[i];
  for (int i = threadIdx.x; i < 1536; i += 256) {
    ((float4*)Ks)[i] = ((const float4*)(kt + base))[i];
    ((float4*)Fs)[i] = ((const float4*)(fo + base))[i];
  }
#endif
  __syncthreads();

  int wid = threadIdx.x >> 5;
  int lane = threadIdx.x & 31;
  int li = lane & 15, hi = lane >> 4, ko = hi * 2;
  int r0 = wid * 16;  // local row base within stripe

  // S stripe = Qstripe * Kt^T with causal mask applied on spill
  for (int j = 0; j < 24; ++j) {
    v8f acc = zero8();
#pragma unroll
    for (int k = 0; k < 4; ++k) {
      v2f a, bb;
      a.x = Qs[(r0 + li) * 16 + 4 * k + ko];
      a.y = Qs[(r0 + li) * 16 + 4 * k + ko + 1];
      bb.x = Ks[(16 * j + li) * 16 + 4 * k + ko];
      bb.y = Ks[(16 * j + li) * 16 + 4 * k + ko + 1];
      acc = wmma4(a, bb, acc);
    }
#pragma unroll
    for (int r = 0; r < 8; ++r) {
      int rl = r0 + r + hi * 8;
      int y = 16 * j + li;
      Ss[rl * 384 + y] = (y <= t0 + rl) ? acc[r] : -3.0e38f;
    }
  }
  __syncthreads();

  if (threadIdx.x < 128) {
    float4* Sr = (float4*)(Ss + threadIdx.x * 384);
    float mx = -3.4e38f;
    for (int i = 0; i < 96; ++i) {
      float4 s4 = Sr[i];
      mx = fmaxf(mx, fmaxf(fmaxf(s4.x, s4.y), fmaxf(s4.z, s4.w)));
    }
    float sum = 0.f;
    for (int i = 0; i < 96; ++i) {
      float4 s4 = Sr[i];
      s4.x = __expf(s4.x - mx); s4.y = __expf(s4.y - mx);
      s4.z = __expf(s4.z - mx); s4.w = __expf(s4.w - mx);
      sum += s4.x + s4.y + s4.z + s4.w;
      Sr[i] = s4;
    }
    rs[threadIdx.x] = 1.0f / sum;
  }
  __syncthreads();

  // t_out stripe = P * Fout  (128x384 @ 384x16)
  v8f acc = zero8();
  for (int k = 0; k < 96; ++k) {
    v2f a, bb;
    a.x = Ss[(r0 + li) * 384 + 4 * k + ko];
    a.y = Ss[(r0 + li) * 384 + 4 * k + ko + 1];
    bb.x = Fs[(4 * k + ko) * 16 + li];
    bb.y = Fs[(4 * k + ko + 1) * 16 + li];
    acc = wmma4(a, bb, acc);
  }
#pragma unroll
  for (int r = 0; r < 8; ++r) {
    int rl = r0 + r + hi * 8;
    to[base + (size_t)(t0 + rl) * DCc + li] = acc[r] * rs[rl];
  }
}

// ---------------------------------------------------------------
// Kernel D: final projection (64x16 GEMM) + bias + PReLU + residual.
// ---------------------------------------------------------------
__global__ __launch_bounds__(256) void proj_kernel(
    const float* __restrict__ ws, const float* __restrict__ x,
    float* __restrict__ out) {
  __shared__ __align__(16) float Wl[64 * 16];
  __shared__ __align__(16) float bl[64];
  __shared__ __align__(16) float al[64];
  for (int i = threadIdx.x; i < 64 * 16; i += 256) Wl[i] = ws[OFF_WP + i];
  if (threadIdx.x < 64) {
    bl[threadIdx.x] = ws[OFF_BP + threadIdx.x];
    al[threadIdx.x] = ws[OFF_AP + threadIdx.x];
  }
  __syncthreads();

  const float* to = ws + OFF_TO;
  int wid = threadIdx.x >> 5;
  int lane = threadIdx.x & 31;
  int li = lane & 15, hi = lane >> 4, ko = hi * 2;
  int b = blockIdx.y;
  int n0 = (blockIdx.x * 8 + wid) * 16;
  const float* tb = to + (size_t)b * Npix * DCc;

  v2f bf[4];
#pragma unroll
  for (int k = 0; k < 4; ++k) {
    bf[k].x = tb[(size_t)(n0 + li) * DCc + 4 * k + ko];
    bf[k].y = tb[(size_t)(n0 + li) * DCc + 4 * k + ko + 1];
  }

#pragma unroll
  for (int m = 0; m < 4; ++m) {
    v8f acc = zero8();
#pragma unroll
    for (int k = 0; k < 4; ++k) {
      v2f a;
      a.x = Wl[(16 * m + li) * 16 + 4 * k + ko];
      a.y = Wl[(16 * m + li) * 16 + 4 * k + ko + 1];
      acc = wmma4(a, bf[k], acc);
    }
    float bi[8], aa[8];
    *(float4*)&bi[0] = *(const float4*)(bl + 16 * m + hi * 8);
    *(float4*)&bi[4] = *(const float4*)(bl + 16 * m + hi * 8 + 4);
    *(float4*)&aa[0] = *(const float4*)(al + 16 * m + hi * 8);
    *(float4*)&aa[4] = *(const float4*)(al + 16 * m + hi * 8 + 4);
#pragma unroll
    for (int r = 0; r < 8; ++r) {
      float y = acc[r] + bi[r];
      y = fmaxf(y, 0.f) + aa[r] * fminf(y, 0.f);
      size_t idx = ((size_t)b * Cc + (16 * m + hi * 8 + r)) * Npix + n0 + li;
      out[idx] = y + x[idx];
    }
  }
}

// ---------------------------------------------------------------
extern "C" void kernel_launch(void* const* d_in, const int* in_sizes, int n_in,
                              void* d_out, int out_size, void* d_ws, size_t ws_size,
                              hipStream_t stream) {
  const float* x       = (const float*)d_in[0];
  const float* fqkv_w  = (const float*)d_in[1];
  const float* fg      = (const float*)d_in[2];
  const float* fbta    = (const float*)d_in[3];
  const float* fm      = (const float*)d_in[4];
  const float* fvv     = (const float*)d_in[5];
  const float* fa      = (const float*)d_in[6];
  const float* tqk_w   = (const float*)d_in[7];
  const float* tg      = (const float*)d_in[8];
  const float* tbta    = (const float*)d_in[9];
  const float* tm      = (const float*)d_in[10];
  const float* tvv     = (const float*)d_in[11];
  const float* ta      = (const float*)d_in[12];
  const float* proj_w  = (const float*)d_in[13];
  const float* pg      = (const float*)d_in[14];
  const float* pbta    = (const float*)d_in[15];
  const float* pm      = (const float*)d_in[16];
  const float* pvv     = (const float*)d_in[17];
  const float* pa      = (const float*)d_in[18];
  float* ws  = (float*)d_ws;
  float* out = (float*)d_out;

  fold_params<<<1, 128, 0, stream>>>(fqkv_w, fg, fbta, fm, fvv, fa,
                                     tqk_w, tg, tbta, tm, tvv, ta,
                                     proj_w, pg, pbta, pm, pvv, pa, ws);

  qkv_kernel<<<dim3(Npix / 16 / 8, Bb), 256, 0, stream>>>(x, ws);

  size_t shB = (size_t)(4096 * 3 + 65536 + 256) * sizeof(float);  // 312320 B
  fattn_kernel<<<dim3(Tt, Bb), 512, shB, stream>>>(ws, ws);

  size_t shC = (size_t)(2048 + 6144 + 6144 + 49152 + 128) * sizeof(float);  // 254464 B
  tattn_kernel<<<dim3(3, Ff, Bb), 256, shC, stream>>>(ws, ws);

  proj_kernel<<<dim3(Npix / 16 / 8, Bb), 256, 0, stream>>>(ws, x, out);
}